// Anchor3_58884001628290
// MI455X (gfx1250) — compile-verified
//
#include <hip/hip_runtime.h>
#include <hip/hip_bf16.h>
#include <cstdint>

// ---------------------------------------------------------------------------
// Sizes (fixed by the reference)
// ---------------------------------------------------------------------------
#define NNODES   1000000
#define NCLS     4096
#define EDIM     64
#define NHEAD    4
#define HDIM     16       // EDIM / NHEAD

typedef __attribute__((ext_vector_type(2))) float v2f;
typedef __attribute__((ext_vector_type(8))) float v8f;

// V_WMMA_F32_16X16X4_F32 : D(16x16 f32) = A(16x4 f32) * B(4x16 f32) + C
// args: (neg_a, A, neg_b, B, c_mod, C, reuse_a, reuse_b)
__device__ __forceinline__ v8f wmma_f32_k4(v8f c, v2f a, v2f b) {
    return __builtin_amdgcn_wmma_f32_16x16x4_f32(
        false, a, false, b, (short)0, c, false, false);
}

// Async global->LDS copy, 16B per lane (GLOBAL_LOAD_ASYNC_TO_LDS_B128, GV
// mode).  IOFFSET applies to BOTH the LDS and global sides (ISA 08 §4.4).
__device__ __forceinline__ void async_copy_b128(uint32_t lds_byte_addr,
                                                const void* gaddr) {
    asm volatile("global_load_async_to_lds_b128 %0, %1, off"
                 :: "v"(lds_byte_addr), "v"(gaddr) : "memory");
}
__device__ __forceinline__ void async_copy_b128_off2048(uint32_t lds_byte_addr,
                                                        const void* gaddr) {
    asm volatile("global_load_async_to_lds_b128 %0, %1, off offset:2048"
                 :: "v"(lds_byte_addr), "v"(gaddr) : "memory");
}
__device__ __forceinline__ void wait_async0() {
    asm volatile("s_wait_asynccnt 0x0" ::: "memory");
}
// low 32 bits of a generic pointer to __shared__ = LDS byte offset
__device__ __forceinline__ uint32_t lds_addr_of(const void* p) {
    return (uint32_t)(uintptr_t)p;
}

// ---------------------------------------------------------------------------
// 1. zero workspace accumulators
// ---------------------------------------------------------------------------
__global__ void zero_kernel(float* __restrict__ p, int n) {
    int i = blockIdx.x * blockDim.x + threadIdx.x;
    if (i < n) p[i] = 0.0f;
}

// ---------------------------------------------------------------------------
// 2. segment sum: sums[cls[node]][e] += feat[node][e]; cnt[cls[node]] += 1
//    one thread per (node, e) element -> fully coalesced loads, atomics hit
//    a 1MB accumulator that stays resident in the 192MB L2.
// ---------------------------------------------------------------------------
__global__ void segsum_kernel(const float* __restrict__ feat,
                              const int* __restrict__ cls,
                              float* __restrict__ sums,
                              float* __restrict__ cnt, int n_elems) {
    int idx = blockIdx.x * blockDim.x + threadIdx.x;
    if (idx >= n_elems) return;
    int node = idx >> 6;
    int e    = idx & 63;
    int c    = cls[node];
    atomicAdd(&sums[c * EDIM + e], feat[idx]);
    if (e == 0) atomicAdd(&cnt[c], 1.0f);
}

// ---------------------------------------------------------------------------
// 3. fea = sums / (cnt + 1e-8)
// ---------------------------------------------------------------------------
__global__ void mean_kernel(const float* __restrict__ sums,
                            const float* __restrict__ cnt,
                            float* __restrict__ fea, int n) {
    int idx = blockIdx.x * blockDim.x + threadIdx.x;
    if (idx >= n) return;
    fea[idx] = sums[idx] / (cnt[idx >> 6] + 1e-8f);
}

// ---------------------------------------------------------------------------
// 4. QKV projection (torch layout: in_w[3E,E] row-major, y = x @ W^T + b)
//    q from sem, k/v from fea.  4096x64 outputs, 64-MAC dots: VALU is ample.
// ---------------------------------------------------------------------------
__global__ void proj_kernel(const float* __restrict__ sem,
                            const float* __restrict__ fea,
                            const float* __restrict__ in_w,
                            const float* __restrict__ in_b,
                            float* __restrict__ qo,
                            float* __restrict__ ko,
                            float* __restrict__ vo, int n) {
    int idx = blockIdx.x * blockDim.x + threadIdx.x;  // n = NCLS*EDIM
    if (idx >= n) return;
    int s = idx >> 6;
    int e = idx & 63;
    const float* srow = sem + s * EDIM;
    const float* frow = fea + s * EDIM;
    const float* wq = in_w + e * EDIM;
    const float* wk = in_w + (EDIM + e) * EDIM;
    const float* wv = in_w + (2 * EDIM + e) * EDIM;
    float aq = in_b[e], ak = in_b[EDIM + e], av = in_b[2 * EDIM + e];
#pragma unroll 8
    for (int j = 0; j < EDIM; ++j) {
        aq += srow[j] * wq[j];
        ak += frow[j] * wk[j];
        av += frow[j] * wv[j];
    }
    qo[idx] = aq; ko[idx] = ak; vo[idx] = av;
}

// ---------------------------------------------------------------------------
// 5. Flash attention, f32 WMMA (16x16x4) + async global->LDS K/V staging.
//    Block = 128 threads = 4 waves; wave w handles head w, one 16-row query
//    tile per block.  Streams 16-key K/V tiles through LDS; online softmax.
// ---------------------------------------------------------------------------
__global__ __launch_bounds__(128)
void attn_kernel(const float* __restrict__ qp, const float* __restrict__ kp,
                 const float* __restrict__ vp, float* __restrict__ attn_out,
                 int T) {
    __shared__ float kt[16][EDIM];         // K tile rows (all heads), 4KB
    __shared__ float vt[16][EDIM];         // V tile rows (all heads), 4KB
    __shared__ float pt[NHEAD][16][17];    // per-wave prob tile (padded)
    __shared__ float alpha_s[NHEAD][16];
    __shared__ float l_s[NHEAD][16];

    const int tid  = threadIdx.x;
    const int wave = tid >> 5;             // == head
    const int lane = tid & 31;
    const int h    = wave;
    const int qb   = blockIdx.x;           // query tile index
    const int n16  = lane & 15;
    const int hi   = lane >> 4;            // lane half
    const int koff = hi * 2;               // A/B fragment K offset

    const uint32_t kt_lds = lds_addr_of(&kt[0][0]) + (uint32_t)tid * 16u;
    const uint32_t vt_lds = lds_addr_of(&vt[0][0]) + (uint32_t)tid * 16u;

    // Q fragments (A-matrix layout) for the 4 K-steps of QK^T, loaded once.
    v2f aq[4];
    {
        const float* qrow = qp + (qb * 16 + n16) * EDIM + h * HDIM;
#pragma unroll
        for (int s = 0; s < 4; ++s) {
            aq[s].x = qrow[s * 4 + koff + 0];
            aq[s].y = qrow[s * 4 + koff + 1];
        }
    }

    v8f  o     = {};
    float m_row = -1e30f;   // valid in lanes 0..15 (row = lane)
    float l_row = 0.0f;

    for (int t0 = 0; t0 < T; t0 += 16) {
        // async K/V tile staging: 16x64 f32 (4KB) each; 128 lanes x 16B
        // covers 2KB per issue, second issue reuses operands with
        // offset:2048 (applies to both LDS and global sides).
        {
            const char* kg = (const char*)(kp + t0 * EDIM) + tid * 16;
            const char* vg = (const char*)(vp + t0 * EDIM) + tid * 16;
            async_copy_b128(kt_lds, kg);
            async_copy_b128_off2048(kt_lds, kg);
            async_copy_b128(vt_lds, vg);
            async_copy_b128_off2048(vt_lds, vg);
            wait_async0();
        }
        __syncthreads();

        // scores tile: C = Q(16x16) * K^T(16x16) via 4 chained WMMA K=4 steps
        v8f c = {};
#pragma unroll
        for (int s = 0; s < 4; ++s) {
            v2f b;  // B[d][n] = K[t0+n][h*16+d]
            b.x = kt[n16][h * HDIM + s * 4 + koff + 0];
            b.y = kt[n16][h * HDIM + s * 4 + koff + 1];
            c = wmma_f32_k4(c, aq[s], b);
        }
        // scale by 1/sqrt(hd)=0.25 and dump to LDS (C layout: VGPR r -> row
        // r + hi*8, col n16)
#pragma unroll
        for (int r = 0; r < 8; ++r)
            pt[wave][r + hi * 8][n16] = c[r] * 0.25f;
        __syncthreads();

        // online softmax: lane L<16 owns query row L of this wave's tile
        if (lane < 16) {
            float tmax = pt[wave][lane][0];
#pragma unroll
            for (int j = 1; j < 16; ++j) tmax = fmaxf(tmax, pt[wave][lane][j]);
            float newm  = fmaxf(m_row, tmax);
            float alpha = __expf(m_row - newm);
            float rsum  = 0.0f;
#pragma unroll
            for (int j = 0; j < 16; ++j) {
                float p = __expf(pt[wave][lane][j] - newm);
                pt[wave][lane][j] = p;
                rsum += p;
            }
            l_row = l_row * alpha + rsum;
            m_row = newm;
            alpha_s[wave][lane] = alpha;
        }
        __syncthreads();

        // rescale running output, then O += P(16x16) * V(16x16)
#pragma unroll
        for (int r = 0; r < 8; ++r) o[r] *= alpha_s[wave][r + hi * 8];
#pragma unroll
        for (int s = 0; s < 4; ++s) {
            v2f a, b;
            a.x = pt[wave][n16][s * 4 + koff + 0];
            a.y = pt[wave][n16][s * 4 + koff + 1];
            b.x = vt[s * 4 + koff + 0][h * HDIM + n16];
            b.y = vt[s * 4 + koff + 1][h * HDIM + n16];
            o = wmma_f32_k4(o, a, b);
        }
        __syncthreads();   // protect kt/vt/pt before next tile
    }

    if (lane < 16) l_s[wave][lane] = l_row;
    __syncthreads();

#pragma unroll
    for (int r = 0; r < 8; ++r) {
        int row = r + hi * 8;
        attn_out[(qb * 16 + row) * EDIM + h * HDIM + n16] = o[r] / l_s[wave][row];
    }
}

// ---------------------------------------------------------------------------
// 6. out projection: final = attn_out @ out_w^T + out_b
// ---------------------------------------------------------------------------
__global__ void outproj_kernel(const float* __restrict__ ain,
                               const float* __restrict__ w,
                               const float* __restrict__ b,
                               float* __restrict__ fin, int n) {
    int idx = blockIdx.x * blockDim.x + threadIdx.x;  // n = NCLS*EDIM
    if (idx >= n) return;
    int s = idx >> 6;
    int e = idx & 63;
    const float* arow = ain + s * EDIM;
    const float* wrow = w + e * EDIM;
    float acc = b[e];
#pragma unroll 8
    for (int j = 0; j < EDIM; ++j) acc += arow[j] * wrow[j];
    fin[idx] = acc;
}

// ---------------------------------------------------------------------------
// 7. gather back to nodes: out[node] = final[cls[node]]   (float4 stores;
//    `final` is 1MB and L2-resident, so this runs at store bandwidth)
// ---------------------------------------------------------------------------
__global__ void scatter_kernel(const float* __restrict__ fin,
                               const int* __restrict__ cls,
                               float* __restrict__ out, int n_nodes) {
    int idx = blockIdx.x * blockDim.x + threadIdx.x;  // n_nodes * 16
    if (idx >= n_nodes * 16) return;
    int node = idx >> 4;
    int q    = idx & 15;
    const float4* f4 = (const float4*)fin;
    float4*       o4 = (float4*)out;
    o4[idx] = f4[cls[node] * 16 + q];
}

// ---------------------------------------------------------------------------
// host-side per-branch pipeline
// ---------------------------------------------------------------------------
static void run_branch(const float* s_feat, const float* sem, const int* cls,
                       const float* in_w, const float* in_b,
                       const float* out_w, const float* out_b,
                       float* out, float* ws, hipStream_t stream) {
    const int FE = NCLS * EDIM;   // 262144
    float* sums = ws;
    float* cnt  = sums + FE;
    float* fea  = cnt + NCLS;
    float* qp   = fea + FE;
    float* kp   = qp + FE;
    float* vp   = kp + FE;
    float* attn = vp + FE;
    float* fin  = attn + FE;

    const int zn = FE + NCLS;
    zero_kernel<<<(zn + 255) / 256, 256, 0, stream>>>(sums, zn);

    const int sn = NNODES * EDIM;
    segsum_kernel<<<(sn + 255) / 256, 256, 0, stream>>>(s_feat, cls, sums, cnt, sn);

    mean_kernel<<<FE / 256, 256, 0, stream>>>(sums, cnt, fea, FE);

    proj_kernel<<<FE / 256, 256, 0, stream>>>(sem, fea, in_w, in_b, qp, kp, vp, FE);

    attn_kernel<<<NCLS / 16, 128, 0, stream>>>(qp, kp, vp, attn, NCLS);

    outproj_kernel<<<FE / 256, 256, 0, stream>>>(attn, out_w, out_b, fin, FE);

    const int gn = NNODES * 16;
    scatter_kernel<<<(gn + 255) / 256, 256, 0, stream>>>(fin, cls, out, NNODES);
}

extern "C" void kernel_launch(void* const* d_in, const int* in_sizes, int n_in,
                              void* d_out, int out_size, void* d_ws, size_t ws_size,
                              hipStream_t stream) {
    const float* v_s     = (const float*)d_in[0];
    const float* c_s     = (const float*)d_in[1];
    const float* v_sem   = (const float*)d_in[2];
    const float* c_sem   = (const float*)d_in[3];
    const int*   v_class = (const int*)d_in[4];
    const int*   c_class = (const int*)d_in[5];
    const float* v_in_w  = (const float*)d_in[6];
    const float* v_in_b  = (const float*)d_in[7];
    const float* v_out_w = (const float*)d_in[8];
    const float* v_out_b = (const float*)d_in[9];
    const float* c_in_w  = (const float*)d_in[10];
    const float* c_in_b  = (const float*)d_in[11];
    const float* c_out_w = (const float*)d_in[12];
    const float* c_out_b = (const float*)d_in[13];

    float* out_v = (float*)d_out;
    float* out_c = out_v + (size_t)NNODES * EDIM;

    float* ws = (float*)d_ws;
    const size_t BR = (size_t)7 * NCLS * EDIM + NCLS;  // floats per branch

    run_branch(v_s, v_sem, v_class, v_in_w, v_in_b, v_out_w, v_out_b,
               out_v, ws, stream);
    run_branch(c_s, c_sem, c_class, c_in_w, c_in_b, c_out_w, c_out_b,
               out_c, ws + BR, stream);
}